// Attention_49340584296826
// MI455X (gfx1250) — compile-verified
//
#include <hip/hip_runtime.h>
#include <hip/hip_bf16.h>

// Shapes from the reference
static constexpr int B   = 32;
static constexpr int TK  = 2048;
static constexpr int N   = 1024;

typedef __attribute__((ext_vector_type(2))) float v2f;
typedef __attribute__((ext_vector_type(8))) float v8f;

// ---------------------------------------------------------------------------
// Kernel 1: dec_fea = s_t_hat @ W_dec^T + b_dec   via V_WMMA_F32_16X16X4_F32
// dec[m,n] = sum_k S[m,k] * W[n,k]   =>  A = S tile (16x4), B[k,n] = W[n,k]
// One wave per 16x16 output tile: 2 m-tiles * 64 n-tiles = 128 waves.
// ---------------------------------------------------------------------------
__global__ __launch_bounds__(32) void dec_proj_wmma(
    const float* __restrict__ S,     // [B, N]  s_t_hat
    const float* __restrict__ W,     // [N, N]  W_dec (row-major, rows = output n)
    const float* __restrict__ bias,  // [N]
    float* __restrict__ dec_fea)     // [B, N]
{
    const int tile  = blockIdx.x;     // 0..127
    const int mtile = tile & 1;       // 2 tiles cover M=32
    const int ntile = tile >> 1;      // 64 tiles cover N=1024
    const int lane  = threadIdx.x;    // 0..31 (wave32)
    const int half  = lane >> 4;      // 0: lanes 0-15, 1: lanes 16-31
    const int l     = lane & 15;
    const int m0 = mtile * 16;
    const int n0 = ntile * 16;

    v8f acc = {};
    for (int k = 0; k < N; k += 4) {
        // A 16x4 f32 layout (ISA 7.12.2): lanes 0-15 -> K = k+0,k+1 ; lanes 16-31 -> K = k+2,k+3
        const float* ap = S + (m0 + l) * N + k + half * 2;
        v2f a; a.x = ap[0]; a.y = ap[1];
        // B 4x16: row K striped across lanes; B[k][n] = W[n][k]
        const float* wp = W + (n0 + l) * N + k + half * 2;
        v2f bfr; bfr.x = wp[0]; bfr.y = wp[1];
        // (neg_a, A, neg_b, B, c_mod, C, reuse_a, reuse_b)
        acc = __builtin_amdgcn_wmma_f32_16x16x4_f32(false, a, false, bfr,
                                                    (short)0, acc, false, false);
    }
    // C/D 16x16 f32 layout: VGPR r, lanes 0-15 -> row r, lanes 16-31 -> row r+8; col = lane%16
    #pragma unroll
    for (int r = 0; r < 8; ++r) {
        const int m = m0 + r + half * 8;
        const int n = n0 + l;
        dec_fea[m * N + n] = acc[r] + bias[n];
    }
}

// ---------------------------------------------------------------------------
// Kernel 2: scores[b,t] = sum_n tanh(ef[b,t,n] + dec_fea[b,n] + cov[b,t]*Wc[n]) * vw[n]
// One wave per (b,t). Streams encoder_feature (268 MB) once, fully coalesced
// float4 loads. dec_fea / Wc / vw are L2-resident.
// ---------------------------------------------------------------------------
__global__ __launch_bounds__(256) void scores_kernel(
    const float* __restrict__ ef,       // [B*TK, N]
    const float* __restrict__ dec_fea,  // [B, N]
    const float* __restrict__ cov,      // [B, TK]
    const float* __restrict__ Wc,       // [N]
    const float* __restrict__ vw,       // [N]
    float* __restrict__ scores)         // [B, TK]
{
    const int wid  = blockIdx.x * 8 + (threadIdx.x >> 5);  // 0..B*TK-1
    const int lane = threadIdx.x & 31;
    const int b    = wid >> 11;   // /TK
    const float c  = cov[wid];

    const float4* efp = (const float4*)(ef + (size_t)wid * N);
    const float4* dfp = (const float4*)(dec_fea + b * N);
    const float4* wcp = (const float4*)Wc;
    const float4* vwp = (const float4*)vw;

    float s = 0.f;
    #pragma unroll
    for (int i = 0; i < 8; ++i) {
        const int idx = lane + i * 32;      // float4 index: covers n = 4*idx..4*idx+3
        const float4 e4 = efp[idx];
        const float4 d4 = dfp[idx];
        const float4 w4 = wcp[idx];
        const float4 v4 = vwp[idx];
        s += tanhf(e4.x + d4.x + c * w4.x) * v4.x;
        s += tanhf(e4.y + d4.y + c * w4.y) * v4.y;
        s += tanhf(e4.z + d4.z + c * w4.z) * v4.z;
        s += tanhf(e4.w + d4.w + c * w4.w) * v4.w;
    }
    #pragma unroll
    for (int off = 16; off; off >>= 1) s += __shfl_xor(s, off, 32);
    if (lane == 0) scores[wid] = s;
}

// ---------------------------------------------------------------------------
// Kernel 3: softmax over T, mask, renormalize, + stmt; coverage_out.
// One block per batch row; 256 threads x 8 elements.
// ---------------------------------------------------------------------------
__global__ __launch_bounds__(256) void softmax_attn_kernel(
    const float* __restrict__ scores,  // [B, TK]
    const float* __restrict__ mask,    // [B, TK]
    const float* __restrict__ stmt,    // [B, TK]
    const float* __restrict__ cov,     // [B, TK]
    float* __restrict__ attn_out,      // [B, TK]
    float* __restrict__ cov_out)       // [B, TK]
{
    __shared__ float red[8];
    const int b    = blockIdx.x;
    const int tid  = threadIdx.x;
    const int lane = tid & 31;
    const int warp = tid >> 5;
    const float* sc = scores + b * TK;

    float v[8], msk[8];
    float m = -3.4e38f;
    #pragma unroll
    for (int i = 0; i < 8; ++i) { v[i] = sc[tid + i * 256]; m = fmaxf(m, v[i]); }
    #pragma unroll
    for (int off = 16; off; off >>= 1) m = fmaxf(m, __shfl_xor(m, off, 32));
    if (lane == 0) red[warp] = m;
    __syncthreads();
    float bm = red[0];
    #pragma unroll
    for (int j = 1; j < 8; ++j) bm = fmaxf(bm, red[j]);
    __syncthreads();

    float sum = 0.f;
    #pragma unroll
    for (int i = 0; i < 8; ++i) { v[i] = __expf(v[i] - bm); sum += v[i]; }
    #pragma unroll
    for (int off = 16; off; off >>= 1) sum += __shfl_xor(sum, off, 32);
    if (lane == 0) red[warp] = sum;
    __syncthreads();
    float tot = 0.f;
    #pragma unroll
    for (int j = 0; j < 8; ++j) tot += red[j];
    __syncthreads();

    // attn_ = softmax * mask, then renormalize by its sum
    const float inv = 1.0f / tot;
    float s2 = 0.f;
    #pragma unroll
    for (int i = 0; i < 8; ++i) {
        msk[i] = mask[b * TK + tid + i * 256];
        v[i] = v[i] * inv * msk[i];
        s2 += v[i];
    }
    #pragma unroll
    for (int off = 16; off; off >>= 1) s2 += __shfl_xor(s2, off, 32);
    if (lane == 0) red[warp] = s2;
    __syncthreads();
    float tot2 = 0.f;
    #pragma unroll
    for (int j = 0; j < 8; ++j) tot2 += red[j];
    const float inv2 = 1.0f / tot2;

    #pragma unroll
    for (int i = 0; i < 8; ++i) {
        const int t = tid + i * 256;
        const float a = v[i] * inv2 + stmt[b * TK + t] * msk[i];
        attn_out[b * TK + t] = a;
        cov_out[b * TK + t]  = cov[b * TK + t] + a;
    }
}

// ---------------------------------------------------------------------------
// Kernel 4: context partials. c_t[b,n] = sum_t attn[b,t]*enc[b,t,n].
// Grid split over (b, n-chunk of 256, t-chunk of 256): 1024 blocks streaming
// encoder_outputs (268 MB) once. attn slice staged in LDS. Deterministic
// partial-sum tree through workspace (no FP atomics).
// ---------------------------------------------------------------------------
__global__ __launch_bounds__(256) void context_partial_kernel(
    const float* __restrict__ enc,   // [B, TK, N]
    const float* __restrict__ attn,  // [B, TK]
    float* __restrict__ part)        // [B, 8, N]
{
    __shared__ float sa[256];
    const int blk = blockIdx.x;        // 0..1023
    const int nch = blk & 3;           // 4 n-chunks of 256
    const int tch = (blk >> 2) & 7;    // 8 t-chunks of 256
    const int b   = blk >> 5;
    const int tid = threadIdx.x;
    const int n   = nch * 256 + tid;

    sa[tid] = attn[b * TK + tch * 256 + tid];
    __syncthreads();

    const float* ep = enc + ((size_t)b * TK + tch * 256) * N + n;
    float acc = 0.f;
    for (int tt = 0; tt < 256; ++tt)
        acc = fmaf(sa[tt], ep[(size_t)tt * N], acc);

    part[((b * 8 + tch) << 10) + n] = acc;
}

__global__ __launch_bounds__(256) void context_reduce_kernel(
    const float* __restrict__ part,  // [B, 8, N]
    float* __restrict__ c_t)         // [B, N]
{
    const int idx = blockIdx.x * 256 + threadIdx.x;  // 0..B*N-1
    const int b = idx >> 10, n = idx & 1023;
    float acc = 0.f;
    #pragma unroll
    for (int tch = 0; tch < 8; ++tch)
        acc += part[((b * 8 + tch) << 10) + n];
    c_t[idx] = acc;
}

// ---------------------------------------------------------------------------
extern "C" void kernel_launch(void* const* d_in, const int* in_sizes, int n_in,
                              void* d_out, int out_size, void* d_ws, size_t ws_size,
                              hipStream_t stream) {
    const float* s_t_hat  = (const float*)d_in[0];  // [B, N]
    const float* enc_out  = (const float*)d_in[1];  // [B, TK, N]
    const float* enc_feat = (const float*)d_in[2];  // [B*TK, N]
    const float* stmt     = (const float*)d_in[3];  // [B, TK]
    const float* mask     = (const float*)d_in[4];  // [B, TK]
    const float* coverage = (const float*)d_in[5];  // [B, TK]
    const float* W_dec    = (const float*)d_in[6];  // [N, N]
    const float* b_dec    = (const float*)d_in[7];  // [N]
    const float* v_w      = (const float*)d_in[8];  // [N]
    const float* W_c      = (const float*)d_in[9];  // [N]

    // Output layout (tuple concat): c_t [B*N], attn_dist [B*TK], coverage_out [B*TK]
    float* c_t      = (float*)d_out;
    float* attn     = c_t + B * N;
    float* cov_out  = attn + B * TK;

    // Workspace layout (rewritten fully every call)
    float* dec_fea = (float*)d_ws;           // B*N      = 32768 floats
    float* scores  = dec_fea + B * N;        // B*TK     = 65536 floats
    float* part    = scores + B * TK;        // B*8*N    = 262144 floats

    dec_proj_wmma<<<128, 32, 0, stream>>>(s_t_hat, W_dec, b_dec, dec_fea);
    scores_kernel<<<(B * TK) / 8, 256, 0, stream>>>(enc_feat, dec_fea, coverage,
                                                    W_c, v_w, scores);
    softmax_attn_kernel<<<B, 256, 0, stream>>>(scores, mask, stmt, coverage,
                                               attn, cov_out);
    context_partial_kernel<<<B * 4 * 8, 256, 0, stream>>>(enc_out, attn, part);
    context_reduce_kernel<<<(B * N) / 256, 256, 0, stream>>>(part, c_t);
}